// PolicyNetwork_64527588655233
// MI455X (gfx1250) — compile-verified
//
#include <hip/hip_runtime.h>
#include <hip/hip_bf16.h>
#include <math.h>

typedef float v2f __attribute__((ext_vector_type(2)));
typedef float v8f __attribute__((ext_vector_type(8)));

#define FDIM 128

// ---------------- utility ----------------
__global__ void zero_f32_kernel(float* __restrict__ p, long long n) {
    long long i = (long long)blockIdx.x * blockDim.x + threadIdx.x;
    if (i < n) p[i] = 0.0f;
}

// ---------------- degree / normalization ----------------
__global__ void deg_count_kernel(const int* __restrict__ dst, unsigned* __restrict__ deg,
                                 long long E) {
    long long e = (long long)blockIdx.x * blockDim.x + threadIdx.x;
    if (e < E) atomicAdd(&deg[dst[e]], 1u);
}

__global__ void make_dinv_kernel(const unsigned* __restrict__ deg, float* __restrict__ dinv,
                                 int N) {
    int i = blockIdx.x * blockDim.x + threadIdx.x;
    // +1 for the self loop -> degree always > 0
    if (i < N) dinv[i] = rsqrtf((float)(deg[i] + 1u));
}

// ---------------- WMMA GEMM: h = x @ W1  (M x 128) * (128 x 128) ----------------
// One wave computes one 16x16 output tile with V_WMMA_F32_16X16X4_F32, K-loop of 32 steps.
// Block = 256 threads = 8 waves -> the 8 N-tiles of one 16-row M-tile.
// A frag (16x4 f32): lane L, VGPR j holds A[L&15][k0 + 2*(L>>4) + j]
// B frag (4x16 f32): lane L, VGPR j holds B[k0 + 2*(L>>4) + j][L&15]
// C/D (16x16 f32) : VGPR r holds C[r + 8*(L>>4)][L&15]
__global__ void gemm_wmma_f32_kernel(const float* __restrict__ A, const float* __restrict__ B,
                                     float* __restrict__ C, int M) {
    const int K = FDIM, NN = FDIM;
    int lane = threadIdx.x & 31;
    int wave = threadIdx.x >> 5;      // N-tile index 0..7
    int tm   = blockIdx.x;            // M-tile index
    int l16  = lane & 15;
    int lhi  = lane >> 4;

    const float* Arow = A + ((size_t)tm * 16 + l16) * K;   // row m = tm*16 + (lane&15)
    const float* Bcol = B + (size_t)(wave * 16 + l16);     // col n = tn*16 + (lane&15)

    v8f c = {};
#pragma unroll 4
    for (int kk = 0; kk < K; kk += 4) {
        int ka = kk + 2 * lhi;
        v2f a, b;
        a.x = Arow[ka + 0];
        a.y = Arow[ka + 1];
        b.x = Bcol[(size_t)(ka + 0) * NN];
        b.y = Bcol[(size_t)(ka + 1) * NN];
        c = __builtin_amdgcn_wmma_f32_16x16x4_f32(false, a, false, b, (short)0, c,
                                                  false, false);
    }

    float* Crow = C + ((size_t)tm * 16 + 8 * lhi) * NN + wave * 16 + l16;
#pragma unroll
    for (int r = 0; r < 8; ++r) Crow[(size_t)r * NN] = c[r];
}

// ---------------- layer-1 edge scatter: agg[d] += dinv[s]*dinv[d] * h[s] ----------------
// One 32-lane wave per edge; each lane moves a float4 (128 floats / edge).
// agg (51.2 MB) is L2-resident -> f32 atomics resolve in L2.
__global__ void scatter_l1_kernel(const float* __restrict__ h, const int* __restrict__ src,
                                  const int* __restrict__ dst, const float* __restrict__ dinv,
                                  float* __restrict__ agg, long long E) {
    long long gid = (long long)blockIdx.x * blockDim.x + threadIdx.x;
    long long e   = gid >> 5;
    int lane      = threadIdx.x & 31;
    if (e >= E) return;
    int s = src[e], d = dst[e];
    float norm = dinv[s] * dinv[d];
    const float4* hs = (const float4*)(h + (size_t)s * FDIM);
    float4 v = hs[lane];
    float* ad = agg + (size_t)d * FDIM + (size_t)lane * 4;
    unsafeAtomicAdd(ad + 0, norm * v.x);
    unsafeAtomicAdd(ad + 1, norm * v.y);
    unsafeAtomicAdd(ad + 2, norm * v.z);
    unsafeAtomicAdd(ad + 3, norm * v.w);
}

// ---------------- layer-1 epilogue: h1 = relu(agg + dinv^2 * h + b1), in place ----------------
__global__ void finalize1_kernel(float* __restrict__ agg, const float* __restrict__ h,
                                 const float* __restrict__ dinv, const float* __restrict__ b1,
                                 long long total) {
    long long i = (long long)blockIdx.x * blockDim.x + threadIdx.x;
    if (i >= total) return;
    int n = (int)(i >> 7);
    int f = (int)(i & (FDIM - 1));
    float di = dinv[n];
    float v  = agg[i] + di * di * h[i] + b1[f];
    agg[i]   = v > 0.0f ? v : 0.0f;
}

// ---------------- z = h1 . W2  (per-node dot of length 128, wave reduce) ----------------
__global__ void dot_w2_kernel(const float* __restrict__ h1, const float* __restrict__ W2,
                              float* __restrict__ z, int N) {
    long long gid = (long long)blockIdx.x * blockDim.x + threadIdx.x;
    long long n   = gid >> 5;
    int lane      = threadIdx.x & 31;
    if (n >= N) return;
    const float4* hp = (const float4*)(h1 + (size_t)n * FDIM);
    const float4* wp = (const float4*)W2;
    float4 a = hp[lane];
    float4 w = wp[lane];
    float s = a.x * w.x + a.y * w.y + a.z * w.z + a.w * w.w;
#pragma unroll
    for (int m = 16; m >= 1; m >>= 1) s += __shfl_xor(s, m, 32);
    if (lane == 0) z[n] = s;
}

// ---------------- layer-2 edge scatter: agg2[d] += dinv[s]*dinv[d] * z[s] ----------------
__global__ void scatter_l2_kernel(const int* __restrict__ src, const int* __restrict__ dst,
                                  const float* __restrict__ dinv, const float* __restrict__ z,
                                  float* __restrict__ agg2, long long E) {
    long long e = (long long)blockIdx.x * blockDim.x + threadIdx.x;
    if (e >= E) return;
    int s = src[e], d = dst[e];
    unsafeAtomicAdd(&agg2[d], dinv[s] * dinv[d] * z[s]);
}

// ---------------- layer-2 epilogue: h2 = agg2 + dinv^2 * z + b2, in place ----------------
__global__ void finalize2_kernel(float* __restrict__ agg2, const float* __restrict__ z,
                                 const float* __restrict__ dinv, const float* __restrict__ b2,
                                 int N) {
    int n = blockIdx.x * blockDim.x + threadIdx.x;
    if (n >= N) return;
    float di = dinv[n];
    agg2[n] = agg2[n] + di * di * z[n] + b2[0];
}

// ---------------- edge probabilities: sigmoid(h2[s] * h2[d]) ----------------
__global__ void edge_prob_kernel(const int* __restrict__ src, const int* __restrict__ dst,
                                 const float* __restrict__ h2, float* __restrict__ out,
                                 long long E) {
    long long e = (long long)blockIdx.x * blockDim.x + threadIdx.x;
    if (e >= E) return;
    float x = h2[src[e]] * h2[dst[e]];
    out[e]  = 1.0f / (1.0f + __expf(-x));
}

extern "C" void kernel_launch(void* const* d_in, const int* in_sizes, int n_in,
                              void* d_out, int out_size, void* d_ws, size_t ws_size,
                              hipStream_t stream) {
    const float* x   = (const float*)d_in[0];   // [N, 128]
    const int*   ei  = (const int*)d_in[1];     // [2, E]
    const float* W1  = (const float*)d_in[2];   // [128, 128]
    const float* b1  = (const float*)d_in[3];   // [128]
    const float* W2  = (const float*)d_in[4];   // [128, 1]
    const float* b2  = (const float*)d_in[5];   // [1]
    float*       out = (float*)d_out;           // [E]

    const int       N = in_sizes[0] / FDIM;     // 100000 (multiple of 16)
    const long long E = (long long)in_sizes[1] / 2;
    const int* src = ei;
    const int* dst = ei + E;

    // workspace layout (floats)
    float*  ws   = (float*)d_ws;
    size_t  N128 = (size_t)N * FDIM;
    float*    h    = ws;                         // N*128  : x @ W1
    float*    agg  = ws + N128;                  // N*128  : layer-1 aggregation -> h1
    float*    dinv = ws + 2 * N128;              // N
    unsigned* deg  = (unsigned*)(ws + 2 * N128 + (size_t)N);  // N
    float*    z    = ws + 2 * N128 + 2 * (size_t)N;           // N : h1 . W2
    float*    agg2 = ws + 2 * N128 + 3 * (size_t)N;           // N : layer-2 agg -> h2

    const int B = 256;
    long long totalNF = (long long)N * FDIM;

    // 1) zero scratch accumulators
    zero_f32_kernel<<<(unsigned)((totalNF + B - 1) / B), B, 0, stream>>>(agg, totalNF);
    zero_f32_kernel<<<(N + B - 1) / B, B, 0, stream>>>((float*)deg, N);
    zero_f32_kernel<<<(N + B - 1) / B, B, 0, stream>>>(agg2, N);

    // 2) degrees (incoming edges), 3) dinv = rsqrt(deg + 1)
    deg_count_kernel<<<(unsigned)((E + B - 1) / B), B, 0, stream>>>(dst, deg, E);
    make_dinv_kernel<<<(N + B - 1) / B, B, 0, stream>>>(deg, dinv, N);

    // 4) h = x @ W1 via fp32 WMMA (M-tiles of 16, block owns a full row of 8 N-tiles)
    gemm_wmma_f32_kernel<<<N / 16, B, 0, stream>>>(x, W1, h, N);

    // 5) layer-1 scatter (wave per edge, float4, L2 f32 atomics)
    {
        long long threads = E * 32;
        scatter_l1_kernel<<<(unsigned)((threads + B - 1) / B), B, 0, stream>>>(
            h, src, dst, dinv, agg, E);
    }

    // 6) h1 = relu(agg + dinv^2*h + b1), in place in agg
    finalize1_kernel<<<(unsigned)((totalNF + B - 1) / B), B, 0, stream>>>(agg, h, dinv, b1,
                                                                          totalNF);

    // 7) z = h1 . W2 (wave per node)
    {
        long long threads = (long long)N * 32;
        dot_w2_kernel<<<(unsigned)((threads + B - 1) / B), B, 0, stream>>>(agg, W2, z, N);
    }

    // 8) layer-2 scalar scatter
    scatter_l2_kernel<<<(unsigned)((E + B - 1) / B), B, 0, stream>>>(src, dst, dinv, z, agg2, E);

    // 9) h2 = agg2 + dinv^2*z + b2, in place in agg2
    finalize2_kernel<<<(N + B - 1) / B, B, 0, stream>>>(agg2, z, dinv, b2, N);

    // 10) edge_probs = sigmoid(h2[src] * h2[dst])
    edge_prob_kernel<<<(unsigned)((E + B - 1) / B), B, 0, stream>>>(src, dst, agg2, out, E);
}